// YOLOXPostProcess_2568390443247
// MI455X (gfx1250) — compile-verified
//
#include <hip/hip_runtime.h>
#include <hip/hip_bf16.h>
#include <stdint.h>

#define IMGF       640.0f
#define NANCH      8400
#define NCLS       80
#define BIMG       32
#define NFLAT      (NANCH * NCLS)   /* 672000 */
#define TOPKC      1024
#define MAXDET     200
#define CANDMAX    4096
#define NBINS      8192
#define SCORE_THRF 0.001f
#define IOU_THRF   0.65f
#define CLASS_OFF  1280.0f          /* 2 * IMG */

typedef __attribute__((ext_vector_type(2))) float v2f;
typedef __attribute__((ext_vector_type(8))) float v8f;

__device__ __forceinline__ unsigned fkey(float f) {
    unsigned u = __float_as_uint(f);
    return (u & 0x80000000u) ? ~u : (u | 0x80000000u);
}

// ---------------------------------------------------------------------------
// Kernel 1: per-image top-1024 selection (histogram radix-cut + bitonic sort)
// followed by box decode of only the selected candidates.
// ---------------------------------------------------------------------------
__global__ __launch_bounds__(1024) void yolox_topk_decode(
    const float* __restrict__ rel, const float* __restrict__ scores,
    const float* __restrict__ anchors,
    float* __restrict__ cs, float* __restrict__ cb, float* __restrict__ cob,
    int* __restrict__ cl, float* __restrict__ ca)
{
    const int b = blockIdx.x;
    const int tid = threadIdx.x;

    __shared__ unsigned long long sh64[CANDMAX];          // 32 KB, re-used
    unsigned* sh = reinterpret_cast<unsigned*>(sh64);     // 8192 u32 view
    __shared__ unsigned chunkS[1024];
    __shared__ unsigned s_thr, s_cnt;

    // ---- phase 1: 8192-bin histogram over monotone keys ----
    for (int i = tid; i < NBINS; i += 1024) sh[i] = 0u;
    __syncthreads();

    const float* base = scores + (size_t)b * NFLAT;
    for (int k = tid; k < NFLAT; k += 1024) {
        if (k + 8192 < NFLAT) __builtin_prefetch(base + k + 8192, 0, 0);
        float f = base[k];
        if (f > SCORE_THRF) atomicAdd(&sh[fkey(f) >> 19], 1u);
    }
    __syncthreads();

    // ---- phase 2: find threshold bin t (largest bin with suffix >= 1024) ----
    unsigned csum = 0;
    for (int q = 0; q < 8; ++q) csum += sh[tid * 8 + q];
    chunkS[tid] = csum;
    __syncthreads();
    if (tid == 0) {
        unsigned acc = 0, t = 0;
        for (int c = 1023; c >= 0; --c) {
            if (acc + chunkS[c] >= TOPKC) {
                for (int bin = c * 8 + 7; bin >= c * 8; --bin) {
                    acc += sh[bin];
                    if (acc >= TOPKC) { t = (unsigned)bin; break; }
                }
                break;
            }
            acc += chunkS[c];
        }
        s_thr = t;
        s_cnt = 0u;
    }
    __syncthreads();
    const unsigned tbin = s_thr;

    // ---- phase 3: compact candidates with bin >= tbin (hist mem reused) ----
    for (int k = tid; k < NFLAT; k += 1024) {
        float f = base[k];
        if (f > SCORE_THRF) {
            unsigned key = fkey(f);
            if ((key >> 19) >= tbin) {
                unsigned pos = atomicAdd(&s_cnt, 1u);
                if (pos < CANDMAX) { sh[pos] = key; sh[CANDMAX + pos] = (unsigned)k; }
            }
        }
    }
    __syncthreads();
    unsigned n = s_cnt; if (n > CANDMAX) n = CANDMAX;

    // ---- phase 4: pad + pack (key desc, idx asc) into u64 in place ----
    unsigned pk[4], pi[4];
    for (int q = 0; q < 4; ++q) {
        int e = tid + q * 1024;
        if ((unsigned)e < n) { pk[q] = sh[e]; pi[q] = sh[CANDMAX + e]; }
        else                 { pk[q] = 0u;   pi[q] = 0xFFFFFFFFu; }
    }
    __syncthreads();
    for (int q = 0; q < 4; ++q) {
        int e = tid + q * 1024;
        sh64[e] = ((unsigned long long)pk[q] << 32) | (unsigned)(~pi[q]);
    }
    __syncthreads();

    // ---- phase 5: bitonic sort, descending, 4096 elements ----
    for (int k2 = 2; k2 <= CANDMAX; k2 <<= 1) {
        for (int j = k2 >> 1; j > 0; j >>= 1) {
            for (int q = 0; q < 4; ++q) {
                int i = tid + q * 1024;
                int l = i ^ j;
                if (l > i) {
                    unsigned long long a = sh64[i], c = sh64[l];
                    bool up = (i & k2) != 0;                 // false => descending run
                    bool sw = up ? (a > c) : (a < c);
                    if (sw) { sh64[i] = c; sh64[l] = a; }
                }
            }
            __syncthreads();
        }
    }

    // ---- phase 6: decode top-1024 candidates ----
    {
        unsigned long long v = sh64[tid];
        unsigned idx = ~((unsigned)v);
        unsigned key = (unsigned)(v >> 32);
        float score, bx0, bx1, bx2, bx3, area;
        int label;
        if (idx == 0xFFFFFFFFu) {
            score = -1.0f; label = 0;
            bx0 = bx1 = bx2 = bx3 = 0.0f; area = 0.0f;
        } else {
            score = __uint_as_float(key ^ 0x80000000u);
            int bi = (int)(idx / NCLS);
            label = (int)(idx % NCLS);
            float a0 = anchors[bi * 4 + 0], a1 = anchors[bi * 4 + 1];
            float a2 = anchors[bi * 4 + 2], a3 = anchors[bi * 4 + 3];
            float ha = a2 - a0, wa = a3 - a1;
            float yca = (a0 + a2) * 0.5f, xca = (a1 + a3) * 0.5f;
            const float* rp = rel + ((size_t)b * NANCH + bi) * 4;
            float ty = rp[0], tx = rp[1], th = rp[2], tw = rp[3];
            float h = expf(th) * ha, w = expf(tw) * wa;
            float yc = ty * ha + yca, xc = tx * wa + xca;
            bx0 = fminf(fmaxf(yc - h * 0.5f, 0.f), IMGF);
            bx1 = fminf(fmaxf(xc - w * 0.5f, 0.f), IMGF);
            bx2 = fminf(fmaxf(yc + h * 0.5f, 0.f), IMGF);
            bx3 = fminf(fmaxf(xc + w * 0.5f, 0.f), IMGF);
            area = fmaxf(bx2 - bx0, 0.f) * fmaxf(bx3 - bx1, 0.f);
        }
        size_t o = (size_t)b * TOPKC + tid;
        cs[o] = score; cl[o] = label; ca[o] = area;
        float off = (float)label * CLASS_OFF;
        cb[o * 4 + 0] = bx0;        cb[o * 4 + 1] = bx1;
        cb[o * 4 + 2] = bx2;        cb[o * 4 + 3] = bx3;
        cob[o * 4 + 0] = bx0 + off; cob[o * 4 + 1] = bx1 + off;
        cob[o * 4 + 2] = bx2 + off; cob[o * 4 + 3] = bx3 + off;
    }
}

// ---------------------------------------------------------------------------
// Kernel 2: 16x16 IoU suppression tiles. Pairwise union term
// (area_i + area_j) computed by v_wmma_f32_16x16x4_f32 as a rank-2 matmul:
//   A[16x4] = [area_i, 1, 0, 0],  B[4x16] = [1; area_j; 0; 0]  =>  D = A*B.
// One wave per tile; row masks gathered with wave32 ballot.
// ---------------------------------------------------------------------------
__global__ __launch_bounds__(256) void yolox_supp_wmma(
    const float* __restrict__ cob, const float* __restrict__ ca,
    unsigned short* __restrict__ supp)
{
    const int lane = threadIdx.x & 31;
    const int tile = blockIdx.x * 8 + (threadIdx.x >> 5);
    const int b = blockIdx.y;
    const int ti = tile >> 6, tj = tile & 63;
    const float* areas = ca + (size_t)b * TOPKC;
    const float* ob = cob + (size_t)b * TOPKC * 4;
    const int i0 = ti * 16, j0 = tj * 16;

    // A 16x4 (MxK): lanes 0-15 hold K=0,1 in v.x,v.y; lanes 16-31 hold K=2,3.
    // B 4x16 (KxN): lanes 0-15 hold rows K=0,1; lanes 16-31 hold rows K=2,3.
    v2f A, B;
    if (lane < 16) {
        A.x = areas[i0 + lane]; A.y = 1.0f;   // A[:,0]=area_i, A[:,1]=1
        B.x = 1.0f;             B.y = areas[j0 + lane];  // B[0,:]=1, B[1,:]=area_j
    } else {
        A.x = 0.0f; A.y = 0.0f; B.x = 0.0f; B.y = 0.0f; // K=2,3 are zero
    }
    v8f C = {0.f, 0.f, 0.f, 0.f, 0.f, 0.f, 0.f, 0.f};
    v8f D = __builtin_amdgcn_wmma_f32_16x16x4_f32(
        false, A, false, B, (short)0, C, false, false);

    const int j = j0 + (lane & 15);
    const float bj0 = ob[j * 4 + 0], bj1 = ob[j * 4 + 1];
    const float bj2 = ob[j * 4 + 2], bj3 = ob[j * 4 + 3];
    const int ihalf = (lane >> 4) << 3;   // lanes 16-31 handle rows r+8
    unsigned myball = 0u;
#pragma unroll
    for (int r = 0; r < 8; ++r) {
        int i = i0 + r + ihalf;
        float bi0 = ob[i * 4 + 0], bi1 = ob[i * 4 + 1];
        float bi2 = ob[i * 4 + 2], bi3 = ob[i * 4 + 3];
        float iy = fminf(bi2, bj2) - fmaxf(bi0, bj0);
        float ix = fminf(bi3, bj3) - fmaxf(bi1, bj1);
        float inter = fmaxf(iy, 0.f) * fmaxf(ix, 0.f);
        float uni = D[r] - inter;                    // area_i + area_j - inter
        float iou = inter / fmaxf(uni, 1e-9f);
        bool pred = (iou > IOU_THRF) && (j > i);
        unsigned bl = (unsigned)__ballot((int)pred); // wave32: low16=row r, high16=row r+8
        if ((lane & 7) == r) myball = bl;
    }
    if (lane < 16) {
        unsigned short m = (lane < 8) ? (unsigned short)(myball & 0xFFFFu)
                                      : (unsigned short)(myball >> 16);
        supp[((size_t)b * TOPKC + (i0 + lane)) * 64 + tj] = m;
    }
}

// ---------------------------------------------------------------------------
// Kernel 3: bitmask greedy NMS (wave 0) + compaction of first 200 kept.
// ---------------------------------------------------------------------------
__global__ __launch_bounds__(1024) void yolox_nms_out(
    const float* __restrict__ cs, const float* __restrict__ cb,
    const int* __restrict__ cl, const unsigned short* __restrict__ supp,
    float* __restrict__ outB, float* __restrict__ outS, float* __restrict__ outL)
{
    const int b = blockIdx.x;
    const int tid = threadIdx.x;
    __shared__ unsigned keep_arr[32];
    __shared__ unsigned scanbuf[1024];

    float sc = cs[(size_t)b * TOPKC + tid];
    bool valid = sc > 0.0f;
    unsigned bl = (unsigned)__ballot((int)valid);
    if ((tid & 31) == 0) keep_arr[tid >> 5] = bl;
    __syncthreads();

    if (tid < 32) {
        unsigned keepw = keep_arr[tid];   // lane w owns keep bits [32w, 32w+32)
        for (int i = 0; i < TOPKC; ++i) {
            unsigned kv = __shfl(keepw, i >> 5);
            if ((kv >> (i & 31)) & 1u) {  // i survives: suppress its row (j>i only)
                const unsigned* row =
                    (const unsigned*)(supp + ((size_t)b * TOPKC + i) * 64);
                keepw &= ~row[tid];
            }
        }
        keep_arr[tid] = keepw;
    }
    __syncthreads();

    unsigned kept = (keep_arr[tid >> 5] >> (tid & 31)) & 1u;
    scanbuf[tid] = kept;
    __syncthreads();
    for (int off = 1; off < 1024; off <<= 1) {
        unsigned add = (tid >= off) ? scanbuf[tid - off] : 0u;
        __syncthreads();
        scanbuf[tid] += add;
        __syncthreads();
    }
    unsigned rank = scanbuf[tid] - kept;      // exclusive rank among kept
    unsigned total = scanbuf[1023];

    if (kept && rank < MAXDET) {
        size_t s = (size_t)b * MAXDET + rank;
        size_t c = (size_t)b * TOPKC + tid;
        outB[s * 4 + 0] = cb[c * 4 + 0];
        outB[s * 4 + 1] = cb[c * 4 + 1];
        outB[s * 4 + 2] = cb[c * 4 + 2];
        outB[s * 4 + 3] = cb[c * 4 + 3];
        outS[s] = sc;
        outL[s] = (float)cl[c];
    }
    if (tid < MAXDET && (unsigned)tid >= total) {   // zero-fill unused slots
        size_t s = (size_t)b * MAXDET + tid;
        outB[s * 4 + 0] = 0.f; outB[s * 4 + 1] = 0.f;
        outB[s * 4 + 2] = 0.f; outB[s * 4 + 3] = 0.f;
        outS[s] = 0.f;
        outL[s] = 0.f;
    }
}

extern "C" void kernel_launch(void* const* d_in, const int* in_sizes, int n_in,
                              void* d_out, int out_size, void* d_ws, size_t ws_size,
                              hipStream_t stream)
{
    (void)in_sizes; (void)n_in; (void)out_size; (void)ws_size;
    const float* rel     = (const float*)d_in[0];   // [32,8400,4]
    const float* scores  = (const float*)d_in[1];   // [32,8400,80]
    const float* anchors = (const float*)d_in[2];   // [8400,4]

    char* ws = (char*)d_ws;
    size_t off = 0;
    float* cs  = (float*)(ws + off); off += (size_t)BIMG * TOPKC * 4;        // scores
    float* cb  = (float*)(ws + off); off += (size_t)BIMG * TOPKC * 16;       // boxes
    float* cob = (float*)(ws + off); off += (size_t)BIMG * TOPKC * 16;       // offset boxes
    int*   cl  = (int*)  (ws + off); off += (size_t)BIMG * TOPKC * 4;        // labels
    float* ca  = (float*)(ws + off); off += (size_t)BIMG * TOPKC * 4;        // areas
    unsigned short* supp = (unsigned short*)(ws + off);                      // [32][1024][64]

    float* outB = (float*)d_out;                 // [32,200,4]
    float* outS = outB + BIMG * MAXDET * 4;      // [32,200]
    float* outL = outS + BIMG * MAXDET;          // [32,200] (labels as float)

    hipLaunchKernelGGL(yolox_topk_decode, dim3(BIMG), dim3(1024), 0, stream,
                       rel, scores, anchors, cs, cb, cob, cl, ca);
    hipLaunchKernelGGL(yolox_supp_wmma, dim3(512, BIMG), dim3(256), 0, stream,
                       cob, ca, supp);
    hipLaunchKernelGGL(yolox_nms_out, dim3(BIMG), dim3(1024), 0, stream,
                       cs, cb, cl, supp, outB, outS, outL);
}